// VarianceAdaptor_73693048865477
// MI455X (gfx1250) — compile-verified
//
#include <hip/hip_runtime.h>
#include <hip/hip_bf16.h>

// Problem constants (match reference file)
#define B_     32
#define TPH    512
#define H_     256
#define NBINS  256
#define MAXLEN 4096   // TPH * DUR_MAX
#define FPB    32     // mel frames per block in the expand kernel

typedef float        v4f __attribute__((ext_vector_type(4)));
typedef unsigned int v4u __attribute__((ext_vector_type(4)));
typedef int          v4i __attribute__((ext_vector_type(4)));
typedef int          v8i __attribute__((ext_vector_type(8)));

#if defined(__has_builtin)
#  if __has_builtin(__builtin_amdgcn_tensor_load_to_lds)
#    define HAVE_TDM 1
#  endif
#endif
#ifndef HAVE_TDM
#  define HAVE_TDM 0
#endif

// ---------------------------------------------------------------------------
// Kernel 1 (prep, tiny): per-batch inclusive cumsum of durations via wave32
// shfl scan, fused with pitch/energy bucketize. Bin tables staged into LDS
// with CDNA5 async global->LDS copies (ASYNCcnt).
// ---------------------------------------------------------------------------
__global__ __launch_bounds__(TPH) void va_prep_kernel(
    const int*   __restrict__ dur,    // (B, TPH)
    const float* __restrict__ ptgt,   // (B, TPH)
    const float* __restrict__ etgt,   // (B, TPH)
    const float* __restrict__ pbins,  // (NBINS-1)
    const float* __restrict__ ebins,  // (NBINS-1)
    int*         __restrict__ csum,   // (B, TPH)
    int*         __restrict__ pidx,   // (B, TPH)
    int*         __restrict__ eidx,   // (B, TPH)
    float*       __restrict__ mel_out)
{
    __shared__ float s_pb[NBINS];
    __shared__ float s_eb[NBINS];
    __shared__ int   s_wsum[16];

    const int b    = blockIdx.x;
    const int tid  = threadIdx.x;      // 0..511
    const int lane = tid & 31;
    const int wid  = tid >> 5;         // 16 waves

    // --- async-stage 255 bin edges per table: 127 x b64 + 1 x b32 ----------
    if (tid < 127) {
        unsigned     dp = (unsigned)(unsigned long long)(&s_pb[tid * 2]);
        unsigned     de = (unsigned)(unsigned long long)(&s_eb[tid * 2]);
        const float* gp = pbins + tid * 2;
        const float* ge = ebins + tid * 2;
        asm volatile("global_load_async_to_lds_b64 %0, %1, off" :: "v"(dp), "v"(gp) : "memory");
        asm volatile("global_load_async_to_lds_b64 %0, %1, off" :: "v"(de), "v"(ge) : "memory");
    } else if (tid == 127) {
        unsigned     dp = (unsigned)(unsigned long long)(&s_pb[254]);
        unsigned     de = (unsigned)(unsigned long long)(&s_eb[254]);
        const float* gp = pbins + 254;
        const float* ge = ebins + 254;
        asm volatile("global_load_async_to_lds_b32 %0, %1, off" :: "v"(dp), "v"(gp) : "memory");
        asm volatile("global_load_async_to_lds_b32 %0, %1, off" :: "v"(de), "v"(ge) : "memory");
    }

    // --- inclusive scan of durations: wave32 shfl scan + wave-sum level ----
    int v = dur[b * TPH + tid];
    #pragma unroll
    for (int off = 1; off < 32; off <<= 1) {
        int n = __shfl_up(v, off, 32);
        if (lane >= off) v += n;
    }
    if (lane == 31) s_wsum[wid] = v;
    asm volatile("s_wait_asynccnt 0" ::: "memory");
    __syncthreads();
    if (tid < 16) {
        int w = s_wsum[tid];
        #pragma unroll
        for (int off = 1; off < 16; off <<= 1) {
            int n = __shfl_up(w, off, 32);
            if (tid >= off) w += n;
        }
        s_wsum[tid] = w;
    }
    __syncthreads();
    const int inc = v + (wid ? s_wsum[wid - 1] : 0);
    csum[b * TPH + tid] = inc;
    if (tid == TPH - 1) mel_out[b] = (float)inc;

    // --- bucketize: searchsorted(bins, v, side='left'), once per phoneme ---
    const float pv = ptgt[b * TPH + tid];
    const float ev = etgt[b * TPH + tid];
    int plo = 0, phi = NBINS - 1;
    while (plo < phi) { int m = (plo + phi) >> 1; if (s_pb[m] < pv) plo = m + 1; else phi = m; }
    int elo = 0, ehi = NBINS - 1;
    while (elo < ehi) { int m = (elo + ehi) >> 1; if (s_eb[m] < ev) elo = m + 1; else ehi = m; }
    pidx[b * TPH + tid] = plo;
    eidx[b * TPH + tid] = elo;
}

// ---------------------------------------------------------------------------
// Kernel 2 (hot): fused embed + length-regulate expansion.
// Grid (MAXLEN/FPB, B), 256 threads. csum[b] staged with the Tensor Data
// Mover (TENSORcnt) when available, else per-lane async-to-LDS (ASYNCcnt).
// One wave resolves ph/pi/ei for the block's 32 frames; 64-lane groups
// stream 3 b128 loads + 1 non-temporal b128 store per frame slice.
// ---------------------------------------------------------------------------
__global__ __launch_bounds__(256) void va_expand_kernel(
    const float* __restrict__ x,      // (B, TPH, H)
    const float* __restrict__ pemb,   // (NBINS, H)
    const float* __restrict__ eemb,   // (NBINS, H)
    const int*   __restrict__ csum,   // (B, TPH)
    const int*   __restrict__ pidx,   // (B, TPH)
    const int*   __restrict__ eidx,   // (B, TPH)
    float*       __restrict__ out)    // (B, MAXLEN, H)
{
    __shared__ int s_csum[TPH];
    __shared__ int s_ph[FPB];
    __shared__ int s_pi[FPB];
    __shared__ int s_ei[FPB];

    const int b      = blockIdx.y;
    const int frame0 = blockIdx.x * FPB;
    const int tid    = threadIdx.x;   // 0..255

#if HAVE_TDM
    // --- TDM: one wave DMAs csum[b, 0:512] (1-D tile, 2 KB) into LDS ------
    if (tid < 32) {
        const unsigned           lds_base = (unsigned)(unsigned long long)(&s_csum[0]);
        const unsigned long long gaddr =
            (unsigned long long)(const void*)(csum + (size_t)b * TPH);
        // D# group 0: count=1 (valid, user), lds_addr, global_addr, type=2
        v4u g0 = { 1u,
                   lds_base,
                   (unsigned)(gaddr & 0xFFFFFFFFu),
                   (unsigned)((gaddr >> 32) & 0x01FFFFFFu) | (2u << 30) };
        // D# group 1: data_size=4B(2), tensor_dim0=512, tensor_dim1=1,
        //             tile_dim0=512, tile_dim1=1, tensor_dim0_stride=512
        v8i g1 = { (int)(2u << 16),     // [17:16] data_size=2 (4 bytes)
                   (int)(512u << 16),   // [63:48] tensor_dim0 lo16 = 512
                   (int)(1u << 16),     // [95:80] tensor_dim1 lo16 = 1
                   (int)(512u << 16),   // [127:112] tile_dim0 = 512
                   (int)1,              // [143:128] tile_dim1 = 1
                   (int)512,            // [191:160] tensor_dim0_stride = 512
                   0, 0 };
        v4i gz = { 0, 0, 0, 0 };
#if __clang_major__ >= 23
        v8i gz8 = { 0, 0, 0, 0, 0, 0, 0, 0 };
        __builtin_amdgcn_tensor_load_to_lds(g0, g1, gz, gz, gz8, 0);
#else
        __builtin_amdgcn_tensor_load_to_lds(g0, g1, gz, gz, 0);
#endif
        __builtin_amdgcn_s_wait_tensorcnt(0);
    }
#else
    // --- fallback: per-lane CDNA5 async copy, 8B per lane (ASYNCcnt) ------
    {
        unsigned   lds_dst = (unsigned)(unsigned long long)(&s_csum[tid * 2]);
        const int* gsrc    = csum + b * TPH + tid * 2;
        asm volatile("global_load_async_to_lds_b64 %0, %1, off"
                     :: "v"(lds_dst), "v"(gsrc) : "memory");
    }
    asm volatile("s_wait_asynccnt 0" ::: "memory");
#endif
    if (tid == 64) {
        __builtin_prefetch(pidx + (size_t)b * TPH, 0, 3);  // global_prefetch_b8
        __builtin_prefetch(eidx + (size_t)b * TPH, 0, 3);
    }
    __syncthreads();

    const int mel_len = s_csum[TPH - 1];

    // One wave resolves per-frame metadata for all FPB frames of this block.
    if (tid < FPB) {
        const int t = frame0 + tid;
        // ph_idx = min(searchsorted(csum, t, side='right'), TPH-1)
        int lo = 0, hi = TPH;
        while (lo < hi) { int mid = (lo + hi) >> 1; if (s_csum[mid] <= t) lo = mid + 1; else hi = mid; }
        const int ph = (lo < TPH) ? lo : (TPH - 1);
        s_ph[tid] = ph;
        s_pi[tid] = pidx[b * TPH + ph];
        s_ei[tid] = eidx[b * TPH + ph];
    }
    __syncthreads();

    const int fidx = tid >> 6;          // frame slot 0..3 per pass
    const int hoff = (tid & 63) << 2;   // float4 offset within H
    #pragma unroll 4
    for (int i = 0; i < FPB / 4; ++i) {
        const int f = i * 4 + fidx;
        const int t = frame0 + f;
        v4f val = {0.0f, 0.0f, 0.0f, 0.0f};
        if (t < mel_len) {
            const int ph = s_ph[f];
            const v4f xv = *(const v4f*)(x    + ((size_t)(b * TPH + ph)) * H_ + hoff);
            const v4f pv = *(const v4f*)(pemb + (size_t)s_pi[f] * H_ + hoff);
            const v4f ev = *(const v4f*)(eemb + (size_t)s_ei[f] * H_ + hoff);
            val = xv + pv + ev;
        }
        // Streaming store: don't let the 134 MB output thrash L2
        __builtin_nontemporal_store(val, (v4f*)(out + ((size_t)b * MAXLEN + t) * H_ + hoff));
    }
}

// ---------------------------------------------------------------------------
extern "C" void kernel_launch(void* const* d_in, const int* in_sizes, int n_in,
                              void* d_out, int out_size, void* d_ws, size_t ws_size,
                              hipStream_t stream) {
    const float* x    = (const float*)d_in[0];
    const float* ptgt = (const float*)d_in[1];
    const float* etgt = (const float*)d_in[2];
    const float* pb   = (const float*)d_in[3];
    const float* eb   = (const float*)d_in[4];
    const float* pe   = (const float*)d_in[5];
    const float* ee   = (const float*)d_in[6];
    const int*   dur  = (const int*)  d_in[7];
    // d_in[8] = max_len scalar (constants baked in)

    float* out = (float*)d_out;
    float* mel = out + (size_t)B_ * MAXLEN * H_;   // mel_len tail (as f32)

    int* cs = (int*)d_ws;                          // 64 KB
    int* pi = cs + (size_t)B_ * TPH;               // 64 KB
    int* ei = pi + (size_t)B_ * TPH;               // 64 KB  (192 KB total)

    va_prep_kernel<<<dim3(B_), dim3(TPH), 0, stream>>>(
        dur, ptgt, etgt, pb, eb, cs, pi, ei, mel);
    va_expand_kernel<<<dim3(MAXLEN / FPB, B_), dim3(256), 0, stream>>>(
        x, pe, ee, cs, pi, ei, out);
}